// GCNOnly_23244363006577
// MI455X (gfx1250) — compile-verified
//
#include <hip/hip_runtime.h>
#include <hip/hip_bf16.h>
#include <math.h>

typedef __attribute__((ext_vector_type(2))) float v2f;
typedef __attribute__((ext_vector_type(8))) float v8f;

__device__ __forceinline__ void atomic_add_f32(float* p, float v) {
    unsafeAtomicAdd(p, v);   // lowers to global_atomic_add_f32 on gfx1250
}

// ---------------- degree / normalization ----------------
__global__ void k_set1(float* __restrict__ p, int n) {
    int i = blockIdx.x * blockDim.x + threadIdx.x;
    if (i < n) p[i] = 1.0f;               // self-loop contribution to degree
}

__global__ void k_deg(const int* __restrict__ dst, float* __restrict__ deg, int E) {
    int e = blockIdx.x * blockDim.x + threadIdx.x;
    if (e < E) atomic_add_f32(&deg[dst[e]], 1.0f);
}

__global__ void k_rsqrt(float* __restrict__ p, int n) {
    int i = blockIdx.x * blockDim.x + threadIdx.x;
    if (i < n) p[i] = rsqrtf(p[i]);       // deg >= 1 always
}

// ---------------- fp32 WMMA GEMM: C[M,NCOL] = A[M,K] @ B[K,NCOL] ----------------
// One wave computes one 16x16 output tile; block covers all NCOL/16 column tiles
// of one 16-row tile. K multiple of 4, NCOL multiple of 16.
template<int K, int NCOL>
__global__ void k_gemm_wmma(const float* __restrict__ A, const float* __restrict__ B,
                            float* __restrict__ C, int M) {
    const int lane = threadIdx.x & 31;
    const int wave = threadIdx.x >> 5;
    const int l16  = lane & 15;
    const int hi   = lane >> 4;            // 0 or 1 (half-wave)
    const int kh   = hi << 1;              // k offset 0 or 2
    const int col  = wave * 16 + l16;

    int row = blockIdx.x * 16 + l16;
    int arow = row < M ? row : (M - 1);
    const float* Ap = A + (size_t)arow * K + kh;

    v8f acc = {0.f, 0.f, 0.f, 0.f, 0.f, 0.f, 0.f, 0.f};
#pragma unroll
    for (int k0 = 0; k0 < K; k0 += 4) {
        v2f a = *(const v2f*)(Ap + k0);                 // A[m][k0+kh .. +1], 8B aligned
        v2f b;
        b.x = B[(size_t)(k0 + kh)     * NCOL + col];    // B[k][n]
        b.y = B[(size_t)(k0 + kh + 1) * NCOL + col];
        acc = __builtin_amdgcn_wmma_f32_16x16x4_f32(
            /*neg_a=*/false, a, /*neg_b=*/false, b,
            /*c_mod=*/(short)0, acc, /*reuse_a=*/false, /*reuse_b=*/false);
    }

    const int rowbase = blockIdx.x * 16 + (hi << 3);    // VGPR r -> row rowbase+r
#pragma unroll
    for (int r = 0; r < 8; ++r) {
        int rr = rowbase + r;
        if (rr < M) C[(size_t)rr * NCOL + col] = acc[r];
    }
}

// ---------------- self-loop init: OUT[i][f] = H[i][f] * dinv[i]^2 ----------------
template<int LOGF>
__global__ void k_selfloop(const float* __restrict__ H, const float* __restrict__ dinv,
                           float* __restrict__ OUT, long long total) {
    long long i = (long long)blockIdx.x * blockDim.x + threadIdx.x;
    if (i < total) {
        int node = (int)(i >> LOGF);
        float dv = dinv[node];
        OUT[i] = H[i] * dv * dv;
    }
}

// ---------------- edge scatter: OUT[dst] += H[src] * dinv[src]*dinv[dst] ---------
// F = 1<<LOGF features; each thread handles one float4 chunk of one edge.
template<int LOGF>
__global__ void k_scatter(const int* __restrict__ src, const int* __restrict__ dst,
                          const float* __restrict__ dinv, const float* __restrict__ H,
                          float* __restrict__ OUT, long long total) {
    long long gid = (long long)blockIdx.x * blockDim.x + threadIdx.x;
    if (gid >= total) return;
    const int QSH = LOGF - 2;                    // quads per edge = F/4
    long long e = gid >> QSH;
    int q = ((int)gid & ((1 << QSH) - 1)) << 2;  // feature offset
    int s = src[e], d = dst[e];
    float norm = dinv[s] * dinv[d];
    const float4 v = *(const float4*)(H + ((size_t)s << LOGF) + q);
    float* o = OUT + ((size_t)d << LOGF) + q;
    atomic_add_f32(o + 0, v.x * norm);
    atomic_add_f32(o + 1, v.y * norm);
    atomic_add_f32(o + 2, v.z * norm);
    atomic_add_f32(o + 3, v.w * norm);
}

// ---------------- layer-1 epilogue: h1 = relu(out1 + b1), F=64 -------------------
__global__ void k_bias_relu64(const float* __restrict__ IN, const float* __restrict__ b,
                              float* __restrict__ OUT, long long total) {
    long long i = (long long)blockIdx.x * blockDim.x + threadIdx.x;
    if (i < total) {
        float v = IN[i] + b[(int)i & 63];
        OUT[i] = v > 0.f ? v : 0.f;
    }
}

// ---------------- head: +b2, log_softmax(32), dot Wd, sigmoid --------------------
__global__ void k_head(const float* __restrict__ AGG, const float* __restrict__ b2,
                       const float* __restrict__ Wd, const float* __restrict__ bd,
                       float* __restrict__ out, int n) {
    int i = blockIdx.x * blockDim.x + threadIdx.x;
    if (i >= n) return;
    const float* row = AGG + (size_t)i * 32;
    float v[32];
    float m = -INFINITY;
#pragma unroll
    for (int j = 0; j < 32; ++j) {
        v[j] = row[j] + b2[j];
        m = fmaxf(m, v[j]);
    }
    float s = 0.f;
#pragma unroll
    for (int j = 0; j < 32; ++j) s += expf(v[j] - m);
    float lse = logf(s);
    float t = bd[0];
#pragma unroll
    for (int j = 0; j < 32; ++j) t += (v[j] - m - lse) * Wd[j];
    out[i] = 1.0f / (1.0f + expf(-t));
}

extern "C" void kernel_launch(void* const* d_in, const int* in_sizes, int n_in,
                              void* d_out, int out_size, void* d_ws, size_t ws_size,
                              hipStream_t stream) {
    const float* x  = (const float*)d_in[0];
    const int*   ei = (const int*)d_in[1];
    const float* W1 = (const float*)d_in[2];
    const float* b1 = (const float*)d_in[3];
    const float* W2 = (const float*)d_in[4];
    const float* b2 = (const float*)d_in[5];
    const float* Wd = (const float*)d_in[6];
    const float* bd = (const float*)d_in[7];

    const int       n = in_sizes[0] / 128;
    const long long E = in_sizes[1] / 2;
    const int* src = ei;
    const int* dst = ei + E;

    // workspace layout (floats): dinv[nAl] | buf1[nAl*64] | buf2[nAl*64]
    float* ws   = (float*)d_ws;
    size_t nAl  = ((size_t)n + 255) & ~(size_t)255;
    float* dinv = ws;
    float* buf1 = ws + nAl;            // H1, later h1
    float* buf2 = buf1 + nAl * 64;     // out1, later H2 | out2
    float* H1   = buf1;
    float* out1 = buf2;
    float* h1   = buf1;                // overwrites H1 after it's consumed
    float* H2   = buf2;                // overwrites out1 after it's consumed
    float* out2 = buf2 + nAl * 32;

    const int T = 256;
    const int nb  = (n + T - 1) / T;
    const int eb  = (int)((E + T - 1) / T);
    const int mt  = (n + 15) / 16;     // 16-row tiles

    // normalization
    k_set1 <<<nb, T, 0, stream>>>(dinv, n);
    k_deg  <<<eb, T, 0, stream>>>(dst, dinv, (int)E);
    k_rsqrt<<<nb, T, 0, stream>>>(dinv, n);

    // layer 1: H1 = X @ W1 ; out1 = selfloop + edge scatter ; h1 = relu(out1+b1)
    k_gemm_wmma<128, 64><<<mt, 128, 0, stream>>>(x, W1, H1, n);
    long long tot1 = (long long)n * 64;
    k_selfloop<6><<<(int)((tot1 + T - 1) / T), T, 0, stream>>>(H1, dinv, out1, tot1);
    long long st1 = E * 16;            // E * (64/4)
    k_scatter<6><<<(int)((st1 + T - 1) / T), T, 0, stream>>>(src, dst, dinv, H1, out1, st1);
    k_bias_relu64<<<(int)((tot1 + T - 1) / T), T, 0, stream>>>(out1, b1, h1, tot1);

    // layer 2: H2 = h1 @ W2 ; out2 = selfloop + edge scatter
    k_gemm_wmma<64, 32><<<mt, 64, 0, stream>>>(h1, W2, H2, n);
    long long tot2 = (long long)n * 32;
    k_selfloop<5><<<(int)((tot2 + T - 1) / T), T, 0, stream>>>(H2, dinv, out2, tot2);
    long long st2 = E * 8;             // E * (32/4)
    k_scatter<5><<<(int)((st2 + T - 1) / T), T, 0, stream>>>(src, dst, dinv, H2, out2, st2);

    // head: +b2, log_softmax, @Wd+bd, sigmoid
    k_head<<<nb, T, 0, stream>>>(out2, b2, Wd, bd, (float*)d_out, n);
}